// GATLayer_48704929137142
// MI455X (gfx1250) — compile-verified
//
#include <hip/hip_runtime.h>
#include <math.h>

typedef float v2f __attribute__((ext_vector_type(2)));
typedef float v8f __attribute__((ext_vector_type(8)));

#define HEADS 4
#define CHAN 32
#define FDIM 128   // F_IN == H*C == 128
#define NEG_SLOPE 0.2f

// ---------------------------------------------------------------------------
// Init: amax = -inf, denom = 0, out accumulator = 0
// ---------------------------------------------------------------------------
__global__ void init_kernel(float* __restrict__ amax, float* __restrict__ denom,
                            float* __restrict__ out, int Nn) {
    int i = blockIdx.x * blockDim.x + threadIdx.x;
    if (i < Nn * HEADS) {
        amax[i]  = -__builtin_inff();
        denom[i] = 0.0f;
    }
    if (i < Nn * FDIM) out[i] = 0.0f;
}

// ---------------------------------------------------------------------------
// GEMM h = x @ W via V_WMMA_F32_16X16X4_F32.
// Wave handles a 16-row x 128-col strip: 8 C-tiles, K-loop of 32 x (K=4).
// A frag (16x4): lanes 0-15 -> M=lane, K={k0,k0+1}; lanes 16-31 -> K={k0+2,k0+3}
// B frag (4x16): lanes 0-15 -> N=lane, K={k0,k0+1}; lanes 16-31 -> K={k0+2,k0+3}
// C/D (16x16):  vgpr r, lane l -> M = (l>=16 ? 8:0)+r, N = l%16
// Requires N % 16 == 0 (true here: 100000).
// ---------------------------------------------------------------------------
__global__ void gemm_wmma(const float* __restrict__ x, const float* __restrict__ W,
                          float* __restrict__ h, int Nn) {
    const int wave = threadIdx.x >> 5;
    const int lane = threadIdx.x & 31;
    const int row0 = (blockIdx.x * 4 + wave) * 16;
    if (row0 >= Nn) return;                       // wave-uniform: EXEC stays full

    v8f acc[8];
#pragma unroll
    for (int t = 0; t < 8; ++t) acc[t] = (v8f)(0.0f);

    const int mrow  = row0 + (lane & 15);
    const int khalf = (lane >> 4) << 1;           // 0 for lanes 0-15, 2 for 16-31
    const int ncol  = lane & 15;

    for (int k0 = 0; k0 < FDIM; k0 += 4) {
        const int ka = k0 + khalf;
        v2f a;
        a.x = x[(size_t)mrow * FDIM + ka];
        a.y = x[(size_t)mrow * FDIM + ka + 1];
#pragma unroll
        for (int t = 0; t < 8; ++t) {
            const int n = t * 16 + ncol;
            v2f b;
            b.x = W[(size_t)ka * FDIM + n];
            b.y = W[(size_t)(ka + 1) * FDIM + n];
            acc[t] = __builtin_amdgcn_wmma_f32_16x16x4_f32(
                false, a, false, b, (short)0, acc[t], false, false);
        }
    }

    const int mbase = row0 + ((lane >> 4) << 3);  // +8 for upper half-wave
#pragma unroll
    for (int t = 0; t < 8; ++t) {
        const int n = t * 16 + ncol;
#pragma unroll
        for (int r = 0; r < 8; ++r)
            h[(size_t)(mbase + r) * FDIM + n] = acc[t][r];
    }
}

// ---------------------------------------------------------------------------
// Per-(node, head) attention terms: a_src = <h, att_src>, a_dst = <h, att_dst>
// ---------------------------------------------------------------------------
__global__ void attn_terms(const float* __restrict__ h,
                           const float* __restrict__ att_src,
                           const float* __restrict__ att_dst,
                           float* __restrict__ a_src, float* __restrict__ a_dst,
                           int Nn) {
    int i = blockIdx.x * blockDim.x + threadIdx.x;
    if (i >= Nn * HEADS) return;
    int node = i >> 2, hd = i & 3;
    const float* hp = h + (size_t)node * FDIM + hd * CHAN;
    const float* as = att_src + hd * CHAN;
    const float* ad = att_dst + hd * CHAN;
    float s = 0.f, d = 0.f;
#pragma unroll
    for (int c = 0; c < CHAN; ++c) {
        float v = hp[c];
        s = fmaf(v, as[c], s);
        d = fmaf(v, ad[c], d);
    }
    a_src[i] = s;
    a_dst[i] = d;
}

__device__ __forceinline__ float leaky(float v) {
    return v > 0.f ? v : NEG_SLOPE * v;
}

// Float atomic-max via monotonic int punning (valid for mixed-sign floats when
// the slot is initialized to -inf = 0xFF800000).
__device__ __forceinline__ void atomicMaxF32(float* addr, float val) {
    if (val >= 0.f)
        atomicMax((int*)addr, __float_as_int(val));
    else
        atomicMin((unsigned int*)addr, __float_as_uint(val));
}

// ---------------------------------------------------------------------------
// Pass 1: segment max of leaky-relu logits over incoming edges of dst
// ---------------------------------------------------------------------------
__global__ void edge_max(const int* __restrict__ ei,
                         const float* __restrict__ a_src,
                         const float* __restrict__ a_dst,
                         float* __restrict__ amax, int Ecnt) {
    int i = blockIdx.x * blockDim.x + threadIdx.x;
    if (i >= Ecnt * HEADS) return;
    int e = i >> 2, hd = i & 3;
    int s = ei[e], d = ei[Ecnt + e];
    float al = leaky(a_src[s * HEADS + hd] + a_dst[d * HEADS + hd]);
    atomicMaxF32(&amax[d * HEADS + hd], al);
}

// ---------------------------------------------------------------------------
// Pass 2: denom = segment sum of exp(logit - amax[dst])
// ---------------------------------------------------------------------------
__global__ void edge_denom(const int* __restrict__ ei,
                           const float* __restrict__ a_src,
                           const float* __restrict__ a_dst,
                           const float* __restrict__ amax,
                           float* __restrict__ denom, int Ecnt) {
    int i = blockIdx.x * blockDim.x + threadIdx.x;
    if (i >= Ecnt * HEADS) return;
    int e = i >> 2, hd = i & 3;
    int s = ei[e], d = ei[Ecnt + e];
    float al = leaky(a_src[s * HEADS + hd] + a_dst[d * HEADS + hd]);
    float mx = amax[d * HEADS + hd];
    if (!__builtin_isfinite(mx)) mx = 0.f;
    atomicAdd(&denom[d * HEADS + hd], expf(al - mx));
}

// ---------------------------------------------------------------------------
// Pass 3: weighted messages, scatter-add into out accumulator.
// One thread per (edge, channel); 128 consecutive threads share an edge so
// h[src*128 + j] gathers and out[dst*128 + j] atomics are fully coalesced;
// h and out are L2-resident (51.2 MB each vs 192 MB L2).
// ---------------------------------------------------------------------------
__global__ void edge_message(const int* __restrict__ ei,
                             const float* __restrict__ a_src,
                             const float* __restrict__ a_dst,
                             const float* __restrict__ amax,
                             const float* __restrict__ denom,
                             const float* __restrict__ h,
                             float* __restrict__ out, int Ecnt) {
    unsigned int i = blockIdx.x * blockDim.x + threadIdx.x;
    unsigned int e = i >> 7;
    if (e >= (unsigned int)Ecnt) return;
    int j  = i & 127;
    int hd = j >> 5;
    int s = ei[e], d = ei[Ecnt + e];
    float al = leaky(a_src[s * HEADS + hd] + a_dst[d * HEADS + hd]);
    float mx = amax[d * HEADS + hd];
    if (!__builtin_isfinite(mx)) mx = 0.f;
    float w = expf(al - mx) / (denom[d * HEADS + hd] + 1e-16f);
    atomicAdd(&out[(size_t)d * FDIM + j], h[(size_t)s * FDIM + j] * w);
}

// ---------------------------------------------------------------------------
// Finalize: out = elu(acc + bias)
// ---------------------------------------------------------------------------
__global__ void finalize(float* __restrict__ out, const float* __restrict__ bias,
                         int total) {
    int i = blockIdx.x * blockDim.x + threadIdx.x;
    if (i >= total) return;
    float v = out[i] + bias[i & (FDIM - 1)];
    out[i] = v > 0.f ? v : expm1f(v);
}

// ---------------------------------------------------------------------------
extern "C" void kernel_launch(void* const* d_in, const int* in_sizes, int n_in,
                              void* d_out, int out_size, void* d_ws, size_t ws_size,
                              hipStream_t stream) {
    const float* x       = (const float*)d_in[0];
    const int*   ei      = (const int*)  d_in[1];
    const float* W       = (const float*)d_in[2];
    const float* att_src = (const float*)d_in[3];
    const float* att_dst = (const float*)d_in[4];
    const float* bias    = (const float*)d_in[5];
    float* out = (float*)d_out;

    const int N = in_sizes[0] / FDIM;
    const int E = in_sizes[1] / 2;

    float* ws     = (float*)d_ws;
    float* h      = ws;                              // N * 128
    float* a_src  = h     + (size_t)N * FDIM;        // N * 4
    float* a_dst  = a_src + (size_t)N * HEADS;       // N * 4
    float* amax   = a_dst + (size_t)N * HEADS;       // N * 4
    float* denom  = amax  + (size_t)N * HEADS;       // N * 4

    const int B = 256;

    init_kernel<<<(N * FDIM + B - 1) / B, B, 0, stream>>>(amax, denom, out, N);

    int tiles  = (N + 15) / 16;
    int gblk   = (tiles + 3) / 4;                    // 4 waves (16-row strips) per block
    gemm_wmma<<<gblk, 128, 0, stream>>>(x, W, h, N);

    attn_terms<<<(N * HEADS + B - 1) / B, B, 0, stream>>>(h, att_src, att_dst,
                                                          a_src, a_dst, N);
    edge_max  <<<(E * HEADS + B - 1) / B, B, 0, stream>>>(ei, a_src, a_dst, amax, E);
    edge_denom<<<(E * HEADS + B - 1) / B, B, 0, stream>>>(ei, a_src, a_dst, amax,
                                                          denom, E);
    long long msg_threads = (long long)E * FDIM;
    edge_message<<<(int)((msg_threads + B - 1) / B), B, 0, stream>>>(
        ei, a_src, a_dst, amax, denom, h, out, E);

    finalize<<<(N * FDIM + B - 1) / B, B, 0, stream>>>(out, bias, N * FDIM);
}